// FixedRadiusGraph_5549097746961
// MI455X (gfx1250) — compile-verified
//
#include <hip/hip_runtime.h>

// CDNA5 / gfx1250 fixed-radius KNN (K=16) using V_WMMA_F32_16X16X4_F32.
//
// d^2(q,c) = |q|^2 - 2 q.c + |c|^2 as a 16x16x4 f32 WMMA:
//   A (16x4, M=candidates): row m = (-2cx, -2cy, -2cz, |c|^2)   [precomputed in LDS]
//   B (4x16, N=queries):    col n = ( qx,   qy,   qz,  1 )
//   C init:                 C[m][n] = |q_n|^2
// f32 WMMA is required: cutoff^2 = 2.5e-3 while |q.c| ~ 3, so f16/bf16
// matrix error (~1e-3 abs) would corrupt the radius test.
//
// Inner loop is software-pipelined in groups of 4 tiles:
//   4x ds_load_b64 (A fragments, issued back-to-back, one dscnt wait)
//   4x v_wmma_f32_16x16x4_f32 (independent: same B, disjoint A/D)
//   4x ~10-VALU min3-tree guard; insert network only on rare hit tiles.

typedef __attribute__((ext_vector_type(2))) float v2f;
typedef __attribute__((ext_vector_type(8))) float v8f;

#define KNN      16
#define CHUNK    256   // candidates staged in LDS per iteration (4 KB as float4)
#define THREADS  256   // 8 wave32 waves
#define QPW      16    // queries per wave (2 lanes per query)

__global__ __launch_bounds__(THREADS)
void frnn_wmma_kernel(const float* __restrict__ pos,
                      const float* __restrict__ cutoff_p,
                      int N,
                      int* __restrict__ nbr_out,
                      int* __restrict__ ctr_out,
                      int* __restrict__ msk_out)
{
    __shared__ float4 cand[CHUNK];   // (-2x, -2y, -2z, |c|^2) per candidate

    const int  lane   = threadIdx.x & 31;
    const int  wave   = threadIdx.x >> 5;
    const int  lane16 = lane & 15;
    const bool hi     = lane >= 16;

    const int qbase = blockIdx.x * (8 * QPW) + wave * QPW;
    const int q     = qbase + lane16;            // this lane's query point

    const float cut = cutoff_p[0];
    const float r2  = cut * cut;

    const float qx = pos[3 * q + 0];
    const float qy = pos[3 * q + 1];
    const float qz = pos[3 * q + 2];
    const float qn2 = qx * qx + qy * qy + qz * qz;

    // B fragment (4x16 f32): v0 = K0 (lanes 0-15) / K2 (lanes 16-31); v1 = K1 / K3.
    v2f bfrag;
    bfrag.x = hi ? qz   : qx;   // K2 : K0
    bfrag.y = hi ? 1.0f : qy;   // K3 : K1

    const v8f acc = { qn2, qn2, qn2, qn2, qn2, qn2, qn2, qn2 };

    // Per-lane LDS base: lanes 0-15 read components (0,1) = (-2x,-2y);
    // lanes 16-31 read components (2,3) = (-2z,|c|^2). One ds_load_b64 per tile.
    const float* lbase = (const float*)cand + (lane16 << 2) + (hi ? 2 : 0);

    // Register-resident sorted neighbor list (ascending d^2, INF padded).
    float dist[KNN];
    int   ind[KNN];
#pragma unroll
    for (int k = 0; k < KNN; ++k) { dist[k] = INFINITY; ind[k] = -1; }

    // Guarded top-K processing of one 16x16 d^2 tile (this lane's 8 rows).
    auto process = [&](const v8f& d2v, int cstart) __attribute__((always_inline)) {
        float m0 = fminf(d2v[0], d2v[1]);
        float m1 = fminf(d2v[2], d2v[3]);
        float m2 = fminf(d2v[4], d2v[5]);
        float m3 = fminf(d2v[6], d2v[7]);
        float vmin = fminf(fminf(m0, m1), fminf(m2, m3));
        bool  hot  = (vmin <= r2) && (vmin < dist[KNN - 1]);
        if (__any(hot)) {
            // C/D layout: lane = query (N), VGPR j = candidate row M
            // (M = j for lanes 0-15, M = j+8 for lanes 16-31).
#pragma unroll
            for (int j = 0; j < 8; ++j) {
                float dd  = d2v[j];
                bool  ins = (dd <= r2) && (dd < dist[KNN - 1]);
                if (__any(ins)) {
                    float di = ins ? dd : INFINITY;   // INF never inserts
                    int   ii = cstart + j;
#pragma unroll
                    for (int p = KNN - 1; p >= 0; --p) {
                        float pd    = (p == 0) ? -INFINITY : dist[p - 1];
                        int   pi    = (p == 0) ? -1        : ind[p - 1];
                        bool  shift = di < pd;                     // p-1 slides down
                        bool  place = (!shift) && (di < dist[p]);  // new lands at p
                        dist[p] = shift ? pd : (place ? di : dist[p]);
                        ind[p]  = shift ? pi : (place ? ii : ind[p]);
                    }
                }
            }
        }
    };

    for (int cb = 0; cb < N; cb += CHUNK) {
        __syncthreads();
        {   // cooperative stage: pre-transform candidates into LDS
            int   ci = cb + (int)threadIdx.x;     // CHUNK == THREADS
            float x = pos[3 * ci + 0];
            float y = pos[3 * ci + 1];
            float z = pos[3 * ci + 2];
            cand[threadIdx.x] =
                make_float4(-2.0f * x, -2.0f * y, -2.0f * z, x * x + y * y + z * z);
            if (cb + CHUNK < N)
                __builtin_prefetch(&pos[3 * (ci + CHUNK)], 0, 3);  // global_prefetch_b8
        }
        __syncthreads();

#pragma unroll 1
        for (int t = 0; t < CHUNK / 16; t += 4) {
            // Issue 4 A-fragment loads back-to-back (constant ds offsets),
            // then 4 independent WMMAs, then the guarded processing.
            v2f a0 = *(const v2f*)(lbase + ((t + 0) << 6));
            v2f a1 = *(const v2f*)(lbase + ((t + 1) << 6));
            v2f a2 = *(const v2f*)(lbase + ((t + 2) << 6));
            v2f a3 = *(const v2f*)(lbase + ((t + 3) << 6));

            v8f d0 = __builtin_amdgcn_wmma_f32_16x16x4_f32(
                false, a0, false, bfrag, (short)0, acc, false, false);
            v8f d1 = __builtin_amdgcn_wmma_f32_16x16x4_f32(
                false, a1, false, bfrag, (short)0, acc, false, false);
            v8f d2 = __builtin_amdgcn_wmma_f32_16x16x4_f32(
                false, a2, false, bfrag, (short)0, acc, false, false);
            v8f d3 = __builtin_amdgcn_wmma_f32_16x16x4_f32(
                false, a3, false, bfrag, (short)0, acc, false, false);

            const int base = cb + t * 16 + (hi ? 8 : 0);
            process(d0, base +  0);
            process(d1, base + 16);
            process(d2, base + 32);
            process(d3, base + 48);
        }
    }

    // Merge the two half-lists of each query (lanes L and L+16) via wave32 shfl.
    float sd[KNN];
    int   si[KNN];
#pragma unroll
    for (int k = 0; k < KNN; ++k) { sd[k] = dist[k]; si[k] = ind[k]; }

#pragma unroll
    for (int k = 0; k < KNN; ++k) {
        float od  = __shfl_xor(sd[k], 16, 32);
        int   oi  = __shfl_xor(si[k], 16, 32);
        bool  ins = od < dist[KNN - 1];
        float di  = ins ? od : INFINITY;
#pragma unroll
        for (int p = KNN - 1; p >= 0; --p) {
            float pd    = (p == 0) ? -INFINITY : dist[p - 1];
            int   pi    = (p == 0) ? -1        : ind[p - 1];
            bool  shift = di < pd;
            bool  place = (!shift) && (di < dist[p]);
            dist[p] = shift ? pd : (place ? di : dist[p]);
            ind[p]  = shift ? pi : (place ? oi : ind[p]);
        }
    }

    // Lanes 0-15 write the final padded (nbr, center, mask) rows.
    if (!hi) {
        const size_t base = (size_t)q * KNN;
#pragma unroll
        for (int k = 0; k < KNN; ++k) {
            bool v = dist[k] <= r2;   // INF-padded entries fail
            nbr_out[base + k] = v ? ind[k] : -1;
            ctr_out[base + k] = v ? q      : -1;
            msk_out[base + k] = v ? 1      : 0;
        }
    }
}

extern "C" void kernel_launch(void* const* d_in, const int* in_sizes, int n_in,
                              void* d_out, int out_size, void* d_ws, size_t ws_size,
                              hipStream_t stream) {
    const float* pos    = (const float*)d_in[0];
    const float* cutoff = (const float*)d_in[1];
    const int    N      = in_sizes[0] / 3;      // 32768

    int* out = (int*)d_out;                     // [nbr | center | mask], each N*16 words
    int* nbr = out;
    int* ctr = out + (size_t)N * KNN;
    int* msk = out + (size_t)2 * N * KNN;

    const int blocks = N / (8 * QPW);           // 128 queries per 256-thread block
    hipLaunchKernelGGL(frnn_wmma_kernel, dim3(blocks), dim3(THREADS), 0, stream,
                       pos, cutoff, N, nbr, ctr, msk);
}